// DAG_RNN_77867757077197
// MI455X (gfx1250) — compile-verified
//
#include <hip/hip_runtime.h>

#define B_ 64
#define T_ 512
#define D_ 128
#define U_ 256
#define G_ (3 * U_)

typedef __attribute__((ext_vector_type(2))) float v2f;
typedef __attribute__((ext_vector_type(8))) float v8f;

// ---------------------------------------------------------------------------
// Kernel A: h[b,u] = (1/T) * sum_{j<i} cond[b,i,j] * H[j,b,u]
// Batched GEMV (no operand reuse across b) -> VALU + prefetch; L2-resident H.
// Summing only j<i exactly matches the reference's zero-initialized H and
// makes the workspace self-initializing (deterministic across graph replays).
// ---------------------------------------------------------------------------
__global__ void __launch_bounds__(U_) state_mix_kernel(
    const float* __restrict__ cond,   // (B,T,T)
    const float* __restrict__ H,      // (T,B,U)
    float* __restrict__ h,            // (B,U)
    int i) {
  const int b = blockIdx.x;
  const int u = threadIdx.x;
  const float* __restrict__ crow = cond + (size_t)b * T_ * T_ + (size_t)i * T_;
  const float* __restrict__ Hp = H + (size_t)b * U_ + u;
  float acc = 0.0f;
  int j = 0;
  for (; j + 4 <= i; j += 4) {
    const float c0 = crow[j + 0], c1 = crow[j + 1];
    const float c2 = crow[j + 2], c3 = crow[j + 3];
    const float h0 = Hp[(size_t)(j + 0) * (B_ * U_)];
    const float h1 = Hp[(size_t)(j + 1) * (B_ * U_)];
    const float h2 = Hp[(size_t)(j + 2) * (B_ * U_)];
    const float h3 = Hp[(size_t)(j + 3) * (B_ * U_)];
    __builtin_prefetch(&Hp[(size_t)(j + 8) * (B_ * U_)], 0, 0);
    acc = fmaf(c0, h0, acc);
    acc = fmaf(c1, h1, acc);
    acc = fmaf(c2, h2, acc);
    acc = fmaf(c3, h3, acc);
  }
  for (; j < i; ++j) acc = fmaf(crow[j], Hp[(size_t)j * (B_ * U_)], acc);
  h[(size_t)b * U_ + u] = acc * (1.0f / (float)T_);
}

// ---------------------------------------------------------------------------
// Async global -> LDS 16-byte copy (per-lane), tracked by ASYNCcnt.
// lds_off = wave-relative LDS byte offset; gaddr = 64-bit global address.
// ---------------------------------------------------------------------------
__device__ __forceinline__ void async_copy_b128(unsigned lds_off,
                                                const float* gaddr) {
  asm volatile("global_load_async_to_lds_b128 %0, %1, off"
               :
               : "v"(lds_off), "v"(gaddr)
               : "memory");
}

__device__ __forceinline__ void async_wait_all() {
  asm volatile("s_wait_asynccnt 0x0" ::: "memory");
}

// ---------------------------------------------------------------------------
// fp32 WMMA 16x16 tile accumulation: acc += A(16xK, LDS) * W(KxG cols col0..+16)
// A layout (ISA 7.12.2, 32-bit A 16x4): lane<16 -> M=lane, K={k,k+1};
// lanes 16..31 -> K={k+2,k+3}.  B symmetric over N.
// ---------------------------------------------------------------------------
template <int K, int PITCH>
__device__ __forceinline__ v8f tile_mm(const float* __restrict__ As,
                                       const float* __restrict__ W,
                                       int col0, v8f acc) {
  const int lane = threadIdx.x & 31;
  const int kh = (lane >> 4) << 1;  // 0 or 2
  const int l = lane & 15;
  const float* __restrict__ arow = As + l * PITCH + kh;
  const float* __restrict__ wp0 = W + (size_t)kh * G_ + col0 + l;
#pragma unroll 4
  for (int k = 0; k < K; k += 4) {
    v2f a, bm;
    a.x = arow[k];
    a.y = arow[k + 1];
    const float* __restrict__ wp = wp0 + (size_t)k * G_;
    bm.x = wp[0];
    bm.y = wp[G_];
    acc = __builtin_amdgcn_wmma_f32_16x16x4_f32(false, a, false, bm,
                                                (short)0, acc, false, false);
  }
  return acc;
}

// ---------------------------------------------------------------------------
// Kernel B: GRU cell for step i.  Block = 16(batch) x 16(u) output tile,
// 4 waves: z-acc, r-acc, xh-acc, hh-acc (xh/hh kept separate because r
// scales only hh).  fp32 WMMA; A tiles staged into LDS via the CDNA5
// async-to-LDS path; weights streamed from L2.
// ---------------------------------------------------------------------------
__global__ void __launch_bounds__(128) gru_cell_kernel(
    const float* __restrict__ x,     // (B,T,D)
    const float* __restrict__ Win,   // (D,G)
    const float* __restrict__ Wrec,  // (U,G)
    const float* __restrict__ bias,  // (2,G)
    const float* __restrict__ hcur,  // (B,U)
    float* __restrict__ Hbuf,        // (T,B,U)
    float* __restrict__ out,         // (B,T,U)
    int i) {
  constexpr int XP = D_ + 4;  // 132 (16B-aligned rows, de-conflicted banks)
  constexpr int HP = U_ + 4;  // 260
  __shared__ __align__(16) float Xs[16 * XP];
  __shared__ __align__(16) float Hs[16 * HP];
  __shared__ float Zs[16][17];
  __shared__ float Rs[16][17];
  __shared__ float XHs[16][17];
  __shared__ float HHs[16][17];

  const int m0 = blockIdx.y * 16;  // batch rows
  const int u0 = blockIdx.x * 16;  // output columns
  const int tid = threadIdx.x;

  // Async-stage x tile (16 x 128) and h tile (16 x 256) into LDS.
  // Trip counts divide evenly by 128 threads -> EXEC all-1s, no divergence.
  const unsigned xs_base = (unsigned)(uintptr_t)Xs;  // LDS byte offset
  const unsigned hs_base = (unsigned)(uintptr_t)Hs;
  for (int t = tid; t < 16 * (D_ / 4); t += 128) {  // 4 iterations
    const int row = t / (D_ / 4), c4 = t % (D_ / 4);
    async_copy_b128(xs_base + (unsigned)(row * XP + c4 * 4) * 4u,
                    x + (size_t)(m0 + row) * T_ * D_ + (size_t)i * D_ + c4 * 4);
  }
  for (int t = tid; t < 16 * (U_ / 4); t += 128) {  // 8 iterations
    const int row = t / (U_ / 4), c4 = t % (U_ / 4);
    async_copy_b128(hs_base + (unsigned)(row * HP + c4 * 4) * 4u,
                    hcur + (size_t)(m0 + row) * U_ + c4 * 4);
  }
  async_wait_all();   // this wave's async copies landed in LDS
  __syncthreads();    // all waves' copies visible

  const int wave = tid >> 5;  // uniform per wave -> EXEC all-1s around WMMA
  const int lane = tid & 31;
  const int l = lane & 15;
  const int rbase = (lane >> 4) << 3;  // C/D rows: lanes>=16 hold M+8
  v8f acc = {0.f, 0.f, 0.f, 0.f, 0.f, 0.f, 0.f, 0.f};
  if (wave == 0) {
    acc = tile_mm<D_, XP>(Xs, Win, 0 * U_ + u0, acc);
    acc = tile_mm<U_, HP>(Hs, Wrec, 0 * U_ + u0, acc);
#pragma unroll
    for (int r = 0; r < 8; ++r) Zs[rbase + r][l] = acc[r];
  } else if (wave == 1) {
    acc = tile_mm<D_, XP>(Xs, Win, 1 * U_ + u0, acc);
    acc = tile_mm<U_, HP>(Hs, Wrec, 1 * U_ + u0, acc);
#pragma unroll
    for (int r = 0; r < 8; ++r) Rs[rbase + r][l] = acc[r];
  } else if (wave == 2) {
    acc = tile_mm<D_, XP>(Xs, Win, 2 * U_ + u0, acc);
#pragma unroll
    for (int r = 0; r < 8; ++r) XHs[rbase + r][l] = acc[r];
  } else {
    acc = tile_mm<U_, HP>(Hs, Wrec, 2 * U_ + u0, acc);
#pragma unroll
    for (int r = 0; r < 8; ++r) HHs[rbase + r][l] = acc[r];
  }
  __syncthreads();

  // Elementwise combine: z = sig(xz+hz), r = sig(xr+hr),
  // hcand = tanh(xh + r*hh), out = z*h + (1-z)*hcand.
#pragma unroll
  for (int e = tid; e < 256; e += 128) {
    const int m = e >> 4, n = e & 15;
    const int col = u0 + n;
    float z = Zs[m][n] + bias[col] + bias[G_ + col];
    float r = Rs[m][n] + bias[U_ + col] + bias[G_ + U_ + col];
    const float xh = XHs[m][n] + bias[2 * U_ + col];
    const float hh = HHs[m][n] + bias[G_ + 2 * U_ + col];
    z = 1.0f / (1.0f + expf(-z));
    r = 1.0f / (1.0f + expf(-r));
    const float hc = tanhf(fmaf(r, hh, xh));
    const float hprev = Hs[m * HP + col];
    const float o = fmaf(z, hprev - hc, hc);  // z*h + (1-z)*hc
    Hbuf[(size_t)i * B_ * U_ + (size_t)(m0 + m) * U_ + col] = o;
    out[(size_t)(m0 + m) * T_ * U_ + (size_t)i * U_ + col] = o;
  }
}

// ---------------------------------------------------------------------------
extern "C" void kernel_launch(void* const* d_in, const int* in_sizes, int n_in,
                              void* d_out, int out_size, void* d_ws,
                              size_t ws_size, hipStream_t stream) {
  (void)in_sizes; (void)n_in; (void)out_size; (void)ws_size;
  const float* inputs = (const float*)d_in[0];  // (B,T,D)
  const float* cond   = (const float*)d_in[1];  // (B,T,T)
  const float* Win    = (const float*)d_in[2];  // (D,3U)
  const float* Wrec   = (const float*)d_in[3];  // (U,3U)
  const float* bias   = (const float*)d_in[4];  // (2,3U)
  float* out = (float*)d_out;                   // (B,T,U)

  float* Hbuf = (float*)d_ws;                        // (T,B,U) = 32 MB
  float* hcur = Hbuf + (size_t)T_ * B_ * U_;         // (B,U)

  for (int i = 0; i < T_; ++i) {
    state_mix_kernel<<<dim3(B_), dim3(U_), 0, stream>>>(cond, Hbuf, hcur, i);
    gru_cell_kernel<<<dim3(U_ / 16, B_ / 16), dim3(128), 0, stream>>>(
        inputs, Win, Wrec, bias, hcur, Hbuf, out, i);
  }
}